// QuanvLayer1D_39642548142546
// MI455X (gfx1250) — compile-verified
//
#include <hip/hip_runtime.h>
#include <hip/hip_bf16.h>

typedef __attribute__((ext_vector_type(2))) float v2f;
typedef __attribute__((ext_vector_type(4))) float v4f;
typedef __attribute__((ext_vector_type(8))) float v8f;

#define NQ 8
#define KERNEL 4
#define LOUT 511
#define LDSTR 264   // padded LDS row stride (floats), 8B-aligned, kills bank conflicts

__device__ __forceinline__ float shflx(float v, int mask) {
    return __shfl_xor(v, mask, 32);
}

// ---- statevector gate engine: 1 wave = 256 complex amps, amp a = r*32 + lane ----
// qubits 0..4 = lane bits (cross-lane shuffles), qubits 5..7 = register bits.

__device__ __forceinline__ void apply1(float vr[8], float vi[8], int q,
                                       float g00r, float g00i, float g01r, float g01i,
                                       float g10r, float g10i, float g11r, float g11i) {
    int lane = threadIdx.x & 31;
    if (q < 5) {
        int mask = 1 << q;
        bool hi = (lane >> q) & 1;
        float c0r = hi ? g10r : g00r, c0i = hi ? g10i : g00i;
        float c1r = hi ? g11r : g01r, c1i = hi ? g11i : g01i;
#pragma unroll
        for (int r = 0; r < 8; ++r) {
            float pr = shflx(vr[r], mask), pi = shflx(vi[r], mask);
            float a0r = hi ? pr : vr[r], a0i = hi ? pi : vi[r];
            float a1r = hi ? vr[r] : pr, a1i = hi ? vi[r] : pi;
            vr[r] = c0r * a0r - c0i * a0i + c1r * a1r - c1i * a1i;
            vi[r] = c0r * a0i + c0i * a0r + c1r * a1i + c1i * a1r;
        }
    } else {
        int h = 1 << (q - 5);
#pragma unroll
        for (int r0 = 0; r0 < 8; ++r0) {
            if (r0 & h) continue;
            int r1 = r0 | h;
            float a0r = vr[r0], a0i = vi[r0], a1r = vr[r1], a1i = vi[r1];
            vr[r0] = g00r * a0r - g00i * a0i + g01r * a1r - g01i * a1i;
            vi[r0] = g00r * a0i + g00i * a0r + g01r * a1i + g01i * a1r;
            vr[r1] = g10r * a0r - g10i * a0i + g11r * a1r - g11i * a1i;
            vi[r1] = g10r * a0i + g10i * a0r + g11r * a1i + g11i * a1r;
        }
    }
}

__device__ __forceinline__ void applyRZ(float vr[8], float vi[8], int q, float ch, float sh) {
    int lane = threadIdx.x & 31;
#pragma unroll
    for (int r = 0; r < 8; ++r) {
        int bit = (q < 5) ? ((lane >> q) & 1) : ((r >> (q - 5)) & 1);
        float ei = bit ? sh : -sh;                 // e^{-+ i t/2}
        float nr = vr[r] * ch - vi[r] * ei;
        float ni = vr[r] * ei + vi[r] * ch;
        vr[r] = nr; vi[r] = ni;
    }
}

__device__ __forceinline__ void applyCNOT(float vr[8], float vi[8], int c, int t) {
    int lane = threadIdx.x & 31;
    if (c < 5 && t < 5) {                          // both lane bits
        int m = 1 << t; bool ctl = (lane >> c) & 1;
#pragma unroll
        for (int r = 0; r < 8; ++r) {
            float pr = shflx(vr[r], m), pi = shflx(vi[r], m);
            vr[r] = ctl ? pr : vr[r];
            vi[r] = ctl ? pi : vi[r];
        }
    } else if (c < 5) {                            // control lane bit, target register bit
        int h = 1 << (t - 5); bool ctl = (lane >> c) & 1;
#pragma unroll
        for (int r0 = 0; r0 < 8; ++r0) {
            if (r0 & h) continue;
            int r1 = r0 | h;
            float t0r = vr[r0], t0i = vi[r0];
            vr[r0] = ctl ? vr[r1] : vr[r0];  vi[r0] = ctl ? vi[r1] : vi[r0];
            vr[r1] = ctl ? t0r : vr[r1];     vi[r1] = ctl ? t0i : vi[r1];
        }
    } else if (t < 5) {                            // control register bit, target lane bit
        int m = 1 << t; int hc = 1 << (c - 5);
#pragma unroll
        for (int r = 0; r < 8; ++r) {
            float pr = shflx(vr[r], m), pi = shflx(vi[r], m);
            if (r & hc) { vr[r] = pr; vi[r] = pi; }
        }
    } else {                                       // both register bits
        int hc = 1 << (c - 5), ht = 1 << (t - 5);
#pragma unroll
        for (int r0 = 0; r0 < 8; ++r0) {
            if (!(r0 & hc) || (r0 & ht)) continue;
            int r1 = r0 | ht;
            float tr = vr[r0], ti = vi[r0];
            vr[r0] = vr[r1]; vi[r0] = vi[r1];
            vr[r1] = tr;     vi[r1] = ti;
        }
    }
}

__device__ __forceinline__ void ansatz(float vr[8], float vi[8], const float* w) {
#pragma unroll
    for (int j = 0; j < NQ; ++j) {
        int cw = (j < NQ - 1) ? j : 0;
        int tw = (j < NQ - 1) ? j + 1 : NQ - 1;
        float t = w[j];
        float ch = __cosf(0.5f * t), sh = __sinf(0.5f * t);
        // RX = [[c, -i s], [-i s, c]]
        apply1(vr, vi, j, ch, 0.f, 0.f, -sh, 0.f, -sh, ch, 0.f);
        applyCNOT(vr, vi, cw, tw);
        // RY = [[c, -s], [s, c]]
        apply1(vr, vi, j, ch, 0.f, -sh, 0.f, sh, 0.f, ch, 0.f);
        applyCNOT(vr, vi, cw, tw);
        applyRZ(vr, vi, j, ch, sh);
    }
}

// ---- kernel 1: build packed B and psi0 = U0|0..0> ----
// B[c][n] = V[n][c] (V = U3*U2*U1), packed so that one b128 load yields
// {Bre[k][n], Bre[k+1][n], Bim[k][n], Bim[k+1][n]} for even k:
//   Bpk[(c>>1)*1024 + n*4 + (c&1)    ] = Bre[c][n]
//   Bpk[(c>>1)*1024 + n*4 + (c&1) + 2] = Bim[c][n]
__global__ __launch_bounds__(256) void build_V(const float* __restrict__ weights,
                                               float* __restrict__ Bpk,
                                               float* __restrict__ psi) {
    int wid = blockIdx.x * 8 + (threadIdx.x >> 5);   // 0..256 jobs
    int lane = threadIdx.x & 31;
    if (wid > 256) return;
    float vr[8], vi[8];
    int init = (wid < 256) ? wid : 0;
#pragma unroll
    for (int r = 0; r < 8; ++r) { vr[r] = (init == r * 32 + lane) ? 1.f : 0.f; vi[r] = 0.f; }
    if (wid < 256) {
        ansatz(vr, vi, weights + 8);
        ansatz(vr, vi, weights + 16);
        ansatz(vr, vi, weights + 24);
        int base = (wid >> 1) * 1024 + (wid & 1);
#pragma unroll
        for (int r = 0; r < 8; ++r) {
            int n = r * 32 + lane;
            Bpk[base + n * 4]     = vr[r];
            Bpk[base + n * 4 + 2] = vi[r];
        }
    } else {
        ansatz(vr, vi, weights + 0);
#pragma unroll
        for (int r = 0; r < 8; ++r) {
            psi[r * 32 + lane]       = vr[r];
            psi[256 + r * 32 + lane] = vi[r];
        }
    }
}

// ---- kernel 2: encode windows (LDS) -> complex WMMA GEMM -> |.|^2 -> <Z_j> ----
__global__ __launch_bounds__(256) void quanv_main(const float* __restrict__ vec,
                                                  const float* __restrict__ Bpk,
                                                  const float* __restrict__ psi,
                                                  float* __restrict__ out,
                                                  int Wtot) {
    __shared__ __align__(16) float sre[16][LDSTR];
    __shared__ __align__(16) float sim[16][LDSTR];
    __shared__ __align__(16) float probs[16][256];

    int wave = threadIdx.x >> 5;
    int lane = threadIdx.x & 31;
    int blk = blockIdx.x;

    // ---- phase 1: encode 2 windows per wave into LDS ----
    for (int it = 0; it < 2; ++it) {
        int m = wave * 2 + it;
        int wg = blk * 16 + m;
        int wgc = (wg < Wtot) ? wg : (Wtot - 1);
        int b = wgc / LOUT, o = wgc % LOUT;
        float vr[8], vi[8];
#pragma unroll
        for (int r = 0; r < 8; ++r) {
            vr[r] = psi[r * 32 + lane];
            vi[r] = psi[256 + r * 32 + lane];
        }
#pragma unroll
        for (int k = 0; k < KERNEL; ++k) {          // RY(window[k]) on qubit k (lane bit)
            int pos = o + k - 1;
            float th = (pos >= 0 && pos < 512) ? vec[b * 512 + pos] : 0.f;
            float ch = __cosf(0.5f * th), sh = __sinf(0.5f * th);
            int mask = 1 << k;
            bool hi = (lane >> k) & 1;
            float c0 = hi ? sh : ch;
            float c1 = hi ? ch : -sh;
#pragma unroll
            for (int r = 0; r < 8; ++r) {
                float pr = shflx(vr[r], mask), pi = shflx(vi[r], mask);
                float a0r = hi ? pr : vr[r], a1r = hi ? vr[r] : pr;
                float a0i = hi ? pi : vi[r], a1i = hi ? vi[r] : pi;
                vr[r] = c0 * a0r + c1 * a1r;
                vi[r] = c0 * a0i + c1 * a1i;
            }
        }
#pragma unroll
        for (int r = 0; r < 8; ++r) {
            sre[m][r * 32 + lane] = vr[r];
            sim[m][r * 32 + lane] = vi[r];
        }
    }
    __syncthreads();

    // ---- phase 2: complex GEMM via V_WMMA_F32_16X16X4_F32 ----
    // out[16 windows][256 amps] = S(16x256) * B(256x256); wave owns N-tiles 2w, 2w+1
    v8f acc_re0 = {}, acc_re1 = {}, acc_im0 = {}, acc_im1 = {};
    int mrow = lane & 15;
    int khalf = (lane >> 4) * 2;
    int n0 = (wave * 2 + 0) * 16 + (lane & 15);
    int n1 = (wave * 2 + 1) * 16 + (lane & 15);
    for (int k = 0; k < 256; k += 4) {
        int k0 = k + khalf;                         // even
        v2f ar = *(const v2f*)&sre[mrow][k0];
        v2f ai = *(const v2f*)&sim[mrow][k0];
        v2f ain; ain.x = -ai.x; ain.y = -ai.y;
        // one b128 per tile: {re_k0, re_k0+1, im_k0, im_k0+1}
        v4f p0 = *(const v4f*)&Bpk[k0 * 512 + n0 * 4];
        v4f p1 = *(const v4f*)&Bpk[k0 * 512 + n1 * 4];
        v2f br0; br0.x = p0.x; br0.y = p0.y;
        v2f bi0; bi0.x = p0.z; bi0.y = p0.w;
        v2f br1; br1.x = p1.x; br1.y = p1.y;
        v2f bi1; bi1.x = p1.z; bi1.y = p1.w;
        // re += Ar*Br - Ai*Bi ; im += Ar*Bi + Ai*Br
        acc_re0 = __builtin_amdgcn_wmma_f32_16x16x4_f32(false, ar,  false, br0, (short)0, acc_re0, false, false);
        acc_re0 = __builtin_amdgcn_wmma_f32_16x16x4_f32(false, ain, false, bi0, (short)0, acc_re0, false, false);
        acc_im0 = __builtin_amdgcn_wmma_f32_16x16x4_f32(false, ar,  false, bi0, (short)0, acc_im0, false, false);
        acc_im0 = __builtin_amdgcn_wmma_f32_16x16x4_f32(false, ai,  false, br0, (short)0, acc_im0, false, false);
        acc_re1 = __builtin_amdgcn_wmma_f32_16x16x4_f32(false, ar,  false, br1, (short)0, acc_re1, false, false);
        acc_re1 = __builtin_amdgcn_wmma_f32_16x16x4_f32(false, ain, false, bi1, (short)0, acc_re1, false, false);
        acc_im1 = __builtin_amdgcn_wmma_f32_16x16x4_f32(false, ar,  false, bi1, (short)0, acc_im1, false, false);
        acc_im1 = __builtin_amdgcn_wmma_f32_16x16x4_f32(false, ai,  false, br1, (short)0, acc_im1, false, false);
    }

    // ---- phase 3: probabilities into LDS ----
    // C/D layout: lane gives N = lane&15; vgpr v gives M = v + 8*(lane>>4)
#pragma unroll
    for (int v = 0; v < 8; ++v) {
        int m = v + 8 * (lane >> 4);
        float re0 = acc_re0[v], im0 = acc_im0[v];
        float re1 = acc_re1[v], im1 = acc_im1[v];
        probs[m][n0] = re0 * re0 + im0 * im0;
        probs[m][n1] = re1 * re1 + im1 * im1;
    }
    __syncthreads();

    // ---- phase 4: signed marginal sums -> <Z_j>, write (bs, 8, 511) ----
    if (threadIdx.x < 128) {
        int m = threadIdx.x >> 3, j = threadIdx.x & 7;
        float ev = 0.f;
        for (int n = 0; n < 256; ++n) {
            float p = probs[m][n];
            ev += ((n >> j) & 1) ? -p : p;
        }
        int wg = blk * 16 + m;
        if (wg < Wtot) {
            int b = wg / LOUT, o = wg % LOUT;
            out[(b * 8 + j) * LOUT + o] = ev;
        }
    }
}

extern "C" void kernel_launch(void* const* d_in, const int* in_sizes, int n_in,
                              void* d_out, int out_size, void* d_ws, size_t ws_size,
                              hipStream_t stream) {
    const float* vec = (const float*)d_in[0];      // (32, 1, 512) f32
    const float* weights = (const float*)d_in[1];  // (4, 8) f32
    float* out = (float*)d_out;                    // (32, 8, 511) f32

    int bs = in_sizes[0] / 512;
    int Wtot = bs * LOUT;

    // workspace layout: Bpk (512KB, packed re/im K-pairs) | psi0 (2KB)
    float* Bpk = (float*)d_ws;
    float* psi = Bpk + 256 * 256 * 2;

    build_V<<<33, 256, 0, stream>>>(weights, Bpk, psi);

    int blocks = (Wtot + 15) / 16;                 // 1022 for bs=32
    quanv_main<<<blocks, 256, 0, stream>>>(vec, Bpk, psi, out, Wtot);
}